// EdgeLoss_66855460929868
// MI455X (gfx1250) — compile-verified
//
#include <hip/hip_runtime.h>
#include <cstdint>

#define IMG_W   256
#define IMG_PIX (IMG_W * IMG_W)
#define N_IMG   128
#define N_TOT   (IMG_PIX * N_IMG)      // 8,388,608 pixels per input
#define TS      32                     // output tile
#define HS      40                     // haloed tile (radius 4 = 2 blur + 1 sobel + 1 NMS)
#define NBLK    (N_IMG * (IMG_W / TS) * (IMG_W / TS))   // 8192 blocks

// 5-tap Gaussian (std=1), normalized
#define G0 0.054488684549643f
#define G1 0.244201342003233f
#define G2 0.402619946894247f

#if defined(__has_builtin)
#  if __has_builtin(__builtin_amdgcn_tensor_load_to_lds)
#    define HAVE_TDM 1
#  endif
#endif
#ifndef HAVE_TDM
#  define HAVE_TDM 0
#endif

typedef __attribute__((ext_vector_type(4))) unsigned int u32x4;
typedef __attribute__((ext_vector_type(8))) int          i32x8;
typedef __attribute__((ext_vector_type(4))) int          i32x4;

// Low 32 bits of a flat pointer into LDS are the LDS byte offset
__device__ __forceinline__ uint32_t lds_off_u32(const void* p) {
  return (uint32_t)(uintptr_t)p;
}

__device__ __forceinline__ void async_ld_b128(uint32_t lds_byte_off, const float* gaddr) {
  // GLOBAL_LOAD_ASYNC_TO_LDS_B128 (GV mode, saddr=off): 16 bytes from vaddr to LDS offset vdst
  asm volatile("global_load_async_to_lds_b128 %0, %1, off"
               :: "v"(lds_byte_off), "v"(gaddr)
               : "memory");
}

__device__ __forceinline__ void wait_async0() {
  asm volatile("s_wait_asynccnt 0x0" ::: "memory");
}

// Issue the 40x40 haloed tile load into LDS buffer `dst`.
// Interior tiles: single TDM 2D strided DMA (packed rows == stride-40 buffer).
// Border tiles: per-lane async B128 with address clamping (masked later in LDS).
__device__ __forceinline__ void issue_tile_load(const float* __restrict__ img,
                                                int ty0, int tx0, int tid,
                                                bool interior, float* dst)
{
#if HAVE_TDM
  if (interior) {
    if (tid == 0) {
      uint64_t ga = (uint64_t)(uintptr_t)(img + (ty0 - 4) * IMG_W + (tx0 - 4));
      u32x4 g0;
      g0[0] = 1u;                                  // count=1, user descriptor
      g0[1] = lds_off_u32(dst);                    // lds_addr (bytes)
      g0[2] = (uint32_t)ga;                        // global_addr[31:0]
      g0[3] = ((uint32_t)(ga >> 32) & 0x01FFFFFFu) // global_addr[56:32]
              | 0x80000000u;                       // type = 2 ("image")
      i32x8 g1;
      g1[0] = (int)(2u << 16);     // data_size = 4 bytes; no mask/pad/iterate
      g1[1] = (int)(256u << 16);   // tensor_dim0 = 256 (low 16 bits)
      g1[2] = (int)(256u << 16);   // tensor_dim0 hi=0 | tensor_dim1 = 256 (low 16)
      g1[3] = (int)(40u << 16);    // tensor_dim1 hi=0 | tile_dim0 = 40
      g1[4] = 40;                  // tile_dim1 = 40, tile_dim2 = 0 (2D)
      g1[5] = 256;                 // tensor_dim0_stride = 256 elements
      g1[6] = 0;                   // stride hi | tensor_dim1_stride lo (unused, 2D)
      g1[7] = 0;
      i32x4 z4 = {0, 0, 0, 0};
#if defined(__clang_major__) && (__clang_major__ >= 23)
      i32x8 z8 = {0, 0, 0, 0, 0, 0, 0, 0};
      __builtin_amdgcn_tensor_load_to_lds(g0, g1, z4, z4, z8, 0);
#else
      __builtin_amdgcn_tensor_load_to_lds(g0, g1, z4, z4, 0);
#endif
    }
    return;
  }
#endif
  const uint32_t base = lds_off_u32(dst);
  for (int c = tid; c < HS * 10; c += 256) {
    int row = c / 10, c4 = c % 10;
    int gy = ty0 - 4 + row;
    int gx = tx0 - 4 + c4 * 4;
    int ei = gy * IMG_W + gx;
    ei = (ei < 0) ? 0 : ((ei > IMG_PIX - 4) ? (IMG_PIX - 4) : ei);  // in-image; masked later
    async_ld_b128(base + (uint32_t)(row * HS + c4 * 4) * 4u, img + ei);
  }
}

// Run the fused Canny pipeline on a preloaded raw 40x40 tile.
// sraw: raw in, becomes mag; t1/t2: scratch.  e[4]: per-thread edge outputs.
__device__ __forceinline__ void process_tile(int ty0, int tx0, int tid,
                                             float* sraw, float* t1, float* t2,
                                             float e[4])
{
  // prep (tanh2sigmoid) + zero outside image (reproduces SAME zero padding)
  for (int i = tid; i < HS * HS; i += 256) {
    int row = i / HS, col = i % HS;
    int gy = ty0 - 4 + row, gx = tx0 - 4 + col;
    bool inb = (gy >= 0) & (gy < IMG_W) & (gx >= 0) & (gx < IMG_W);
    float v = sraw[i];
    sraw[i] = inb ? (v + 1.0f) * 0.5f : 0.0f;
  }
  __syncthreads();

  // horizontal 5-tap blur: rows 0..39, cols 2..37 (mask OOB columns)
  for (int j = tid; j < HS * 36; j += 256) {
    int row = j / 36, col = 2 + j % 36;
    const float* r = sraw + row * HS + col;
    float h = G0 * (r[-2] + r[2]) + G1 * (r[-1] + r[1]) + G2 * r[0];
    int gx = tx0 - 4 + col;
    t1[row * HS + col] = (gx >= 0 && gx < IMG_W) ? h : 0.0f;
  }
  __syncthreads();

  // vertical 5-tap blur: rows 2..37, cols 2..37 (mask OOB rows)
  for (int j = tid; j < 36 * 36; j += 256) {
    int row = 2 + j / 36, col = 2 + j % 36;
    const float* p = t1 + row * HS + col;
    float v = G0 * (p[-2 * HS] + p[2 * HS]) + G1 * (p[-HS] + p[HS]) + G2 * p[0];
    int gy = ty0 - 4 + row;
    t2[row * HS + col] = (gy >= 0 && gy < IMG_W) ? v : 0.0f;
  }
  __syncthreads();

  // Sobel + gradient magnitude: rows 3..36, cols 3..36; mag overwrites raw (dead)
  for (int j = tid; j < 34 * 34; j += 256) {
    int row = 3 + j / 34, col = 3 + j % 34;
    const float* p = t2 + row * HS + col;
    float a = p[-HS - 1], b = p[-HS], c = p[-HS + 1];
    float d = p[-1],                  f = p[1];
    float g = p[HS - 1],  h = p[HS],  i2 = p[HS + 1];
    float gxv = (a - c) + 2.0f * (d - f) + (g - i2);
    float gyv = (a + 2.0f * b + c) - (g + 2.0f * h + i2);
    float m = sqrtf(gxv * gxv + gyv * gyv);
    int gy = ty0 - 4 + row, gx = tx0 - 4 + col;
    bool inb = (gy >= 0) & (gy < IMG_W) & (gx >= 0) & (gx < IMG_W);
    sraw[row * HS + col] = inb ? m : 0.0f;
  }
  __syncthreads();

  // directional NMS + threshold on the 32x32 outputs (rows/cols 4..35).
  // Orientation bin b = round(atan2(gy,gx)/45deg) mod 8 via octant compares;
  // idx_neg = b (offset +d), idx_pos = b+4 (offset -d).
  const int drt[8] = {0, 1, 1, 1, 0, -1, -1, -1};
  const int dct[8] = {1, 1, 0, -1, -1, -1, 0, 1};
  for (int q = 0; q < 4; ++q) {
    int j = tid + q * 256;
    int oy = j >> 5, ox = j & 31;
    int row = 4 + oy, col = 4 + ox;
    const float* p = t2 + row * HS + col;
    float a = p[-HS - 1], b = p[-HS], c = p[-HS + 1];
    float d = p[-1],                  f = p[1];
    float g = p[HS - 1],  h = p[HS],  i2 = p[HS + 1];
    float gxv = (a - c) + 2.0f * (d - f) + (g - i2);
    float gyv = (a + 2.0f * b + c) - (g + 2.0f * h + i2);
    float ax = fabsf(gxv), ay = fabsf(gyv);
    int bidx;
    if (ay <= 0.41421357f * ax)       bidx = (gxv >= 0.0f) ? 0 : 4;  // ~0 / 180 deg
    else if (ay >= 2.4142134f * ax)   bidx = (gyv >= 0.0f) ? 2 : 6;  // ~90 / 270 deg
    else if (gxv > 0.0f)              bidx = (gyv > 0.0f) ? 1 : 7;
    else                              bidx = (gyv > 0.0f) ? 3 : 5;
    int dr = drt[bidx], dc = dct[bidx];
    float m  = sraw[row * HS + col];
    float nn = sraw[(row + dr) * HS + (col + dc)];
    float np = sraw[(row - dr) * HS + (col - dc)];
    float sel = fminf(m - nn, m - np);
    e[q] = (sel > 0.0f && m >= 2.0f) ? m : 0.0f;
  }
  __syncthreads();   // scratch buffers free for next image
}

// ---- main fused kernel: per-tile L1 partial sums -------------------------
__global__ __launch_bounds__(256)
void canny_tile_kernel(const float* __restrict__ gt, const float* __restrict__ pred,
                       float* __restrict__ partial)
{
  __shared__ float sA[HS * HS];   // pred raw -> pred mag
  __shared__ float sB[HS * HS];   // gt   raw -> gt   mag
  __shared__ float t1[HS * HS];   // h-blur scratch
  __shared__ float t2[HS * HS];   // v-blur scratch
  __shared__ float sred[8];

  int tid  = threadIdx.x;
  int blk  = blockIdx.x;
  int img  = blk >> 6;
  int tile = blk & 63;
  int ty0  = (tile >> 3) * TS;
  int tx0  = (tile & 7) * TS;
  bool interior = (tx0 > 0) && (tx0 < IMG_W - TS) && (ty0 > 0) && (ty0 < IMG_W - TS);
  const float* gbase = gt   + (size_t)img * IMG_PIX;
  const float* pbase = pred + (size_t)img * IMG_PIX;

  // Issue both loads up front; both complete before any compute starts.
  issue_tile_load(pbase, ty0, tx0, tid, interior, sA);
  issue_tile_load(gbase, ty0, tx0, tid, interior, sB);
  wait_async0();
#if HAVE_TDM
  __builtin_amdgcn_s_wait_tensorcnt(0);   // no-op for waves that issued none
#endif
  __syncthreads();

  float ep[4], eg[4];
  process_tile(ty0, tx0, tid, sA, t1, t2, ep);
  process_tile(ty0, tx0, tid, sB, t1, t2, eg);

  float acc = 0.0f;
  #pragma unroll
  for (int q = 0; q < 4; ++q) acc += fabsf(ep[q] - eg[q]);

  // wave32 reduction, then across the 8 waves
  #pragma unroll
  for (int o = 16; o > 0; o >>= 1) acc += __shfl_xor(acc, o, 32);
  if ((tid & 31) == 0) sred[tid >> 5] = acc;
  __syncthreads();
  if (tid == 0) {
    float t = 0.0f;
    #pragma unroll
    for (int w = 0; w < 8; ++w) t += sred[w];
    partial[blk] = t;
  }
}

// ---- deterministic final reduction ---------------------------------------
__global__ __launch_bounds__(256)
void canny_reduce_kernel(const float* __restrict__ partial, float* __restrict__ out)
{
  __shared__ float sred[8];
  int tid = threadIdx.x;
  float acc = 0.0f;
  for (int i = tid; i < NBLK; i += 256) acc += partial[i];
  #pragma unroll
  for (int o = 16; o > 0; o >>= 1) acc += __shfl_xor(acc, o, 32);
  if ((tid & 31) == 0) sred[tid >> 5] = acc;
  __syncthreads();
  if (tid == 0) {
    float t = 0.0f;
    #pragma unroll
    for (int w = 0; w < 8; ++w) t += sred[w];
    out[0] = t * (1.0f / (float)N_TOT);
  }
}

extern "C" void kernel_launch(void* const* d_in, const int* in_sizes, int n_in,
                              void* d_out, int out_size, void* d_ws, size_t ws_size,
                              hipStream_t stream) {
  const float* gt   = (const float*)d_in[0];   // data_input
  const float* pred = (const float*)d_in[1];   // model_output
  float* out     = (float*)d_out;
  float* partial = (float*)d_ws;               // 8192 floats = 32 KB scratch

  canny_tile_kernel<<<NBLK, 256, 0, stream>>>(gt, pred, partial);
  canny_reduce_kernel<<<1, 256, 0, stream>>>(partial, out);
}